// RGCN_7318624272991
// MI455X (gfx1250) — compile-verified
//
#include <hip/hip_runtime.h>

// ---------------------------------------------------------------------------
// RGCN (2-layer, R relations, 128-dim) for gfx1250 / MI455X.
// fp32 WMMA (v_wmma_f32_16x16x4_f32) for the dense transforms; L2-resident
// gather + atomic-scatter for edge aggregation; degree norms computed once;
// bias folded into accumulator init; ReLU templated into the layer-2 A-load.
// ---------------------------------------------------------------------------

typedef __attribute__((ext_vector_type(2))) float v2f;
typedef __attribute__((ext_vector_type(8))) float v8f;

#define FDIM 128

// --- zero a float buffer ----------------------------------------------------
__global__ __launch_bounds__(256) void rgcn_zero_kernel(float* __restrict__ p, int n) {
    int i = blockIdx.x * blockDim.x + threadIdx.x;
    if (i < n) p[i] = 0.0f;
}

// --- count degrees: one thread per (relation, edge) -------------------------
__global__ __launch_bounds__(256) void rgcn_degree_kernel(
    const int* __restrict__ src, const int* __restrict__ dst,
    float* __restrict__ deg_out, float* __restrict__ deg_in,
    int E, int N, int RE) {
    int t = blockIdx.x * blockDim.x + threadIdx.x;
    if (t >= RE) return;
    int r = t / E;
    atomicAdd(&deg_out[(size_t)r * N + src[t]], 1.0f);
    atomicAdd(&deg_in [(size_t)r * N + dst[t]], 1.0f);
}

// --- deg -> rsqrt(max(deg,1)) in place --------------------------------------
__global__ __launch_bounds__(256) void rgcn_rsqrt_kernel(float* __restrict__ p, int n) {
    int i = blockIdx.x * blockDim.x + threadIdx.x;
    if (i < n) p[i] = rsqrtf(fmaxf(p[i], 1.0f));
}

// --- init accumulator rows to sum_r b[r][:] ---------------------------------
__global__ __launch_bounds__(256) void rgcn_bias_init_kernel(
    float* __restrict__ out, const float* __restrict__ b, int R, int total) {
    int i = blockIdx.x * blockDim.x + threadIdx.x;
    if (i >= total) return;
    int c = i & (FDIM - 1);
    float v = 0.0f;
    for (int r = 0; r < R; ++r) v += b[r * FDIM + c];
    out[i] = v;
}

// --- WMMA GEMM: M = (scale .* act(X)) @ W,  X:[N,128], W:[128,128] ----------
// One block = 256 threads = 8 waves; block handles 16 rows, wave w handles
// columns [16w, 16w+16).  K-loop: 32 chained v_wmma_f32_16x16x4_f32.
// EXEC all-ones inside (uniform block-level early-out only), as WMMA requires.
// RELU is a compile-time parameter so layer 1 carries zero select overhead.
template <bool RELU>
__global__ __launch_bounds__(256) void rgcn_gemm_wmma_kernel(
    const float* __restrict__ X, const float* __restrict__ W,
    const float* __restrict__ rsq, float* __restrict__ M, int n) {
    int row0 = blockIdx.x * 16;
    if (row0 >= n) return;                    // uniform across the block
    int wave = threadIdx.x >> 5;              // 0..7 -> 16-col slab
    int lane = threadIdx.x & 31;
    int lh   = lane >> 4;                     // lane half: K-pair select
    int lm   = lane & 15;                     // row (A) / col (B,D) index
    int col0 = wave * 16;

    int arow = row0 + lm;
    float scale = rsq[arow];
    const float* xrow = X + (size_t)arow * FDIM;

    v8f acc = {};
#pragma unroll 4
    for (int kb = 0; kb < FDIM; kb += 4) {
        int k0 = kb + 2 * lh;                 // this lane's K pair {k0, k0+1}
        // A fragment (16x4 f32): lanes 0-15 hold K={kb,kb+1}, lanes 16-31
        // K={kb+2,kb+3}.  k0 is even and rows are 512B-aligned -> one b64 load.
        v2f a = *(const v2f*)(xrow + k0);
        if (RELU) {
            a.x = fmaxf(a.x, 0.0f);
            a.y = fmaxf(a.y, 0.0f);
        }
        v2f A = { a.x * scale, a.y * scale };
        // B fragment (4x16 f32): lane lm = column, VGPR j = K k0+j
        v2f B = { W[(size_t)(k0 + 0) * FDIM + col0 + lm],
                  W[(size_t)(k0 + 1) * FDIM + col0 + lm] };
        // 8 args: (neg_a, A, neg_b, B, c_mod, C, reuse_a, reuse_b)
        acc = __builtin_amdgcn_wmma_f32_16x16x4_f32(
            false, A, false, B, (short)0, acc, false, false);
    }
    // D layout (16x16 f32): VGPR v -> row = v + 8*lh, col = lm
    float* mrow = M + (size_t)row0 * FDIM + col0 + lm;
#pragma unroll
    for (int v = 0; v < 8; ++v)
        mrow[(size_t)(v + 8 * lh) * FDIM] = acc[v];
}

// --- edge scatter: out[dst] += m[src] * rsq_in[dst] -------------------------
// One wave per edge; each lane moves a float4 (coalesced 512B gather per edge)
// and does 4 global f32 atomics.  m and out are L2-resident (51.2 MB each).
__global__ __launch_bounds__(256) void rgcn_scatter_kernel(
    const float* __restrict__ M, const int* __restrict__ src,
    const int* __restrict__ dst, const float* __restrict__ rsq_in,
    float* __restrict__ out, int E) {
    int t = blockIdx.x * blockDim.x + threadIdx.x;
    int e = t >> 5;
    if (e >= E) return;
    int lane = t & 31;
    int s = src[e];
    int d = dst[e];
    float sc = rsq_in[d];
    const float4* mp = (const float4*)(M + (size_t)s * FDIM) + lane;
    float4 v = *mp;
    float* op = out + (size_t)d * FDIM + lane * 4;
    atomicAdd(op + 0, v.x * sc);
    atomicAdd(op + 1, v.y * sc);
    atomicAdd(op + 2, v.z * sc);
    atomicAdd(op + 3, v.w * sc);
}

extern "C" void kernel_launch(void* const* d_in, const int* in_sizes, int n_in,
                              void* d_out, int out_size, void* d_ws, size_t ws_size,
                              hipStream_t stream) {
    const float* x   = (const float*)d_in[0];
    const float* W1  = (const float*)d_in[1];
    const float* b1  = (const float*)d_in[2];
    const float* W2  = (const float*)d_in[3];
    const float* b2  = (const float*)d_in[4];
    const int*   src = (const int*)d_in[5];
    const int*   dst = (const int*)d_in[6];

    const int N = in_sizes[0] / FDIM;
    const int R = in_sizes[1] / (FDIM * FDIM);
    const int E = in_sizes[5] / R;

    float* ws      = (float*)d_ws;
    float* rsq_out = ws;                                    // [R*N]
    float* rsq_in  = rsq_out + (size_t)R * N;               // [R*N]
    float* h       = rsq_in  + (size_t)R * N;               // [N*128] layer-1 acc
    float* m       = h       + (size_t)N * FDIM;            // [N*128] per-rel GEMM out
    float* out     = (float*)d_out;                         // [N*128]

    const int TB = 256;
    const int degTotal = 2 * R * N;
    const int RE       = R * E;
    const int feats    = N * FDIM;

    // 1) degree norms (shared by both layers)
    rgcn_zero_kernel<<<(degTotal + TB - 1) / TB, TB, 0, stream>>>(rsq_out, degTotal);
    rgcn_degree_kernel<<<(RE + TB - 1) / TB, TB, 0, stream>>>(
        src, dst, rsq_out, rsq_in, E, N, RE);
    rgcn_rsqrt_kernel<<<(degTotal + TB - 1) / TB, TB, 0, stream>>>(rsq_out, degTotal);

    const int gemmGrid    = (N + 15) / 16;
    const int scatterGrid = (int)(((size_t)E * 32 + TB - 1) / TB);

    // 2) layer 1: h = sum_r [ scatter_r( (rsq_out_r .* x) @ W1_r ) ] + sum_r b1_r
    rgcn_bias_init_kernel<<<(feats + TB - 1) / TB, TB, 0, stream>>>(h, b1, R, feats);
    for (int r = 0; r < R; ++r) {
        rgcn_gemm_wmma_kernel<false><<<gemmGrid, TB, 0, stream>>>(
            x, W1 + (size_t)r * FDIM * FDIM, rsq_out + (size_t)r * N, m, N);
        rgcn_scatter_kernel<<<scatterGrid, TB, 0, stream>>>(
            m, src + (size_t)r * E, dst + (size_t)r * E, rsq_in + (size_t)r * N, h, E);
    }

    // 3) layer 2: out = sum_r [ scatter_r( (rsq_out_r .* relu(h)) @ W2_r ) ] + sum_r b2_r
    rgcn_bias_init_kernel<<<(feats + TB - 1) / TB, TB, 0, stream>>>(out, b2, R, feats);
    for (int r = 0; r < R; ++r) {
        rgcn_gemm_wmma_kernel<true><<<gemmGrid, TB, 0, stream>>>(
            h, W2 + (size_t)r * FDIM * FDIM, rsq_out + (size_t)r * N, m, N);
        rgcn_scatter_kernel<<<scatterGrid, TB, 0, stream>>>(
            m, src + (size_t)r * E, dst + (size_t)r * E, rsq_in + (size_t)r * N, out, E);
    }
}